// SMPLlayer_38637525795077
// MI455X (gfx1250) — compile-verified
//
#include <hip/hip_runtime.h>
#include <math.h>

#define B_N   512
#define V_N   6890
#define J_N   24
#define NSH   10
#define NPB   207
#define NCOL  (V_N * 3)        // 20670
#define KPAD  220              // [shapes(10) | 0,0 | pose_feature(207) | 0]
#define NT_GROUPS  323         // ceil(ceil(20670/16)/4)
#define NM_GROUPS  16          // 512 / 32 rows per wave

typedef __attribute__((ext_vector_type(2))) float v2f;
typedef __attribute__((ext_vector_type(8))) float v8f;

__device__ const int c_parents[24] = {-1,0,0,0,1,2,3,4,5,6,7,8,9,9,9,12,13,14,16,17,18,19,20,21};

// ---------------------------------------------------------------------------
// Kernel 1: project J_regressor onto {shapedirs (l=0..9), v_template (l=10)}
// jsmall[j*33 + d*11 + l] = sum_v Jreg[j,v] * (l<10 ? shapedirs[v,d,l] : v_template[v,d])
// ---------------------------------------------------------------------------
__global__ void k_jsmall(const float* __restrict__ jreg, const float* __restrict__ sdirs,
                         const float* __restrict__ vtmpl, float* __restrict__ jsmall) {
  const int o = blockIdx.x;               // 0..791
  const int j = o / 33, rem = o % 33, d = rem / 11, l = rem % 11;
  const int tid = threadIdx.x;
  float s = 0.0f;
  for (int v = tid; v < V_N; v += 256) {
    const float x = (l < 10) ? sdirs[(v * 3 + d) * NSH + l] : vtmpl[v * 3 + d];
    s += jreg[j * V_N + v] * x;
  }
  __shared__ float red[256];
  red[tid] = s;
  __syncthreads();
  for (int st = 128; st > 0; st >>= 1) {
    if (tid < st) red[tid] += red[tid + st];
    __syncthreads();
  }
  if (tid == 0) jsmall[o] = red[0];
}

// ---------------------------------------------------------------------------
// Kernel 2: transpose weights (V,24) -> (24,V) for coalesced LBS reads
// ---------------------------------------------------------------------------
__global__ void k_wtrans(const float* __restrict__ w, float* __restrict__ wt) {
  const int i = blockIdx.x * 256 + threadIdx.x;
  if (i < V_N * J_N) {
    const int v = i / J_N, j = i % J_N;
    wt[j * V_N + v] = w[i];
  }
}

// ---------------------------------------------------------------------------
// Kernel 2b: transpose shapedirs (V*3, 10) -> sdT (12, V*3), rows 10..11 = 0
// ---------------------------------------------------------------------------
__global__ void k_sdt(const float* __restrict__ sdirs, float* __restrict__ sdt) {
  const int i = blockIdx.x * 256 + threadIdx.x;
  if (i < 12 * NCOL) {
    const int k = i / NCOL, n = i % NCOL;
    sdt[i] = (k < NSH) ? sdirs[n * NSH + k] : 0.0f;
  }
}

// ---------------------------------------------------------------------------
// Kernel 3: per-batch rodrigues + kinematic chain.
// Writes the packed A matrix row apack[b][0..219] = [shapes | 0,0 | pf | 0]
// and folded per-joint transforms mt (B,24,12):
//   mt[0..8] = rot(Rh) @ Rg_j ; mt[9..11] = rot(Rh) @ (tg_j - Rg_j@Jrest_j) + Th
// ---------------------------------------------------------------------------
__device__ __forceinline__ void rodrigues9(float x, float y, float z, float* R) {
  x += 1e-8f; y += 1e-8f; z += 1e-8f;
  const float ang = sqrtf(x * x + y * y + z * z);
  const float inv = 1.0f / ang;
  const float rx = x * inv, ry = y * inv, rz = z * inv;
  const float c = cosf(ang), s = sinf(ang), t = 1.0f - c;
  R[0] = c + t * rx * rx;      R[1] = t * rx * ry - s * rz; R[2] = t * rx * rz + s * ry;
  R[3] = t * rx * ry + s * rz; R[4] = c + t * ry * ry;      R[5] = t * ry * rz - s * rx;
  R[6] = t * rx * rz - s * ry; R[7] = t * ry * rz + s * rx; R[8] = c + t * rz * rz;
}

__global__ void k_chain(const float* __restrict__ poses, const float* __restrict__ shapes,
                        const float* __restrict__ Rh, const float* __restrict__ Th,
                        const float* __restrict__ jsmall,
                        float* __restrict__ apack, float* __restrict__ mt) {
  const int b = blockIdx.x;
  const int j = threadIdx.x;               // blockDim = 32
  __shared__ float sR[24][9];
  __shared__ float sJ[24][3];
  __shared__ float sG[24][12];
  __shared__ float ssh[10];
  float* arow = apack + (size_t)b * KPAD;
  if (j < 10) { const float sv = shapes[b * NSH + j]; ssh[j] = sv; arow[j] = sv; }
  if (j == 10 || j == 11) arow[j] = 0.0f;
  if (j == 0) arow[KPAD - 1] = 0.0f;
  __syncthreads();
  if (j < 24) {
    float R[9];
    rodrigues9(poses[b * 72 + j * 3], poses[b * 72 + j * 3 + 1], poses[b * 72 + j * 3 + 2], R);
#pragma unroll
    for (int c = 0; c < 9; ++c) sR[j][c] = R[c];
    if (j >= 1) {                          // pose_feature -> apack[12 + (j-1)*9 + c]
#pragma unroll
      for (int c = 0; c < 9; ++c)
        arow[12 + (j - 1) * 9 + c] = R[c] - ((c % 4 == 0) ? 1.0f : 0.0f);
    }
#pragma unroll
    for (int d = 0; d < 3; ++d) {
      const float* js = jsmall + j * 33 + d * 11;
      float s = js[10];
#pragma unroll
      for (int l = 0; l < 10; ++l) s += ssh[l] * js[l];
      sJ[j][d] = s;
    }
  }
  __syncthreads();
  if (j == 0) {                            // sequential chain (23 small matmuls)
#pragma unroll
    for (int c = 0; c < 9; ++c) sG[0][c] = sR[0][c];
    sG[0][9] = sJ[0][0]; sG[0][10] = sJ[0][1]; sG[0][11] = sJ[0][2];
    for (int i = 1; i < 24; ++i) {
      const int p = c_parents[i];
      const float rel0 = sJ[i][0] - sJ[p][0];
      const float rel1 = sJ[i][1] - sJ[p][1];
      const float rel2 = sJ[i][2] - sJ[p][2];
      for (int r = 0; r < 3; ++r) {
        const float g0 = sG[p][r * 3 + 0], g1 = sG[p][r * 3 + 1], g2 = sG[p][r * 3 + 2];
        for (int cc = 0; cc < 3; ++cc)
          sG[i][r * 3 + cc] = g0 * sR[i][cc] + g1 * sR[i][3 + cc] + g2 * sR[i][6 + cc];
        sG[i][9 + r] = sG[p][9 + r] + g0 * rel0 + g1 * rel1 + g2 * rel2;
      }
    }
  }
  __syncthreads();
  if (j < 24) {
    float G[12];
#pragma unroll
    for (int c = 0; c < 12; ++c) G[c] = sG[j][c];
    const float jx = sJ[j][0], jy = sJ[j][1], jz = sJ[j][2];
    const float t0 = G[9]  - (G[0] * jx + G[1] * jy + G[2] * jz);
    const float t1 = G[10] - (G[3] * jx + G[4] * jy + G[5] * jz);
    const float t2 = G[11] - (G[6] * jx + G[7] * jy + G[8] * jz);
    float Rr[9];
    rodrigues9(Rh[b * 3], Rh[b * 3 + 1], Rh[b * 3 + 2], Rr);
    const float th[3] = {Th[b * 3], Th[b * 3 + 1], Th[b * 3 + 2]};
    float* o = mt + ((size_t)b * 24 + j) * 12;
#pragma unroll
    for (int r = 0; r < 3; ++r) {
      const float a0 = Rr[r * 3 + 0], a1 = Rr[r * 3 + 1], a2 = Rr[r * 3 + 2];
#pragma unroll
      for (int cc = 0; cc < 3; ++cc)
        o[r * 3 + cc] = a0 * G[cc] + a1 * G[3 + cc] + a2 * G[6 + cc];
      o[9 + r] = a0 * t0 + a1 * t1 + a2 * t2 + th[r];
    }
  }
}

// ---------------------------------------------------------------------------
// Kernel 4: fused GEMM via fp32 WMMA 16x16x4, branchless inner loops:
//   v_posed[b, n] = v_template[n] + apack (512x220) @ [sdT | posedirs] (220x20670)
// Each wave: 2 M-tiles (32 batch rows) x 4 N-tiles (64 cols) -> 8 accumulators;
// B fragments are loaded once and reused by both M-tiles (8 B loads + 2 A loads
// per 8 WMMAs). Out-of-range columns are clamped (those lanes' accumulators are
// never stored). The K tail step is peeled; the posedirs row-207 overrun there
// is clamped and killed by the zero A column at k=219.
// ---------------------------------------------------------------------------
__global__ void __launch_bounds__(256) k_vposed(const float* __restrict__ apack,
                                                const float* __restrict__ sdt,
                                                const float* __restrict__ pdirs,
                                                const float* __restrict__ vtmpl,
                                                float* __restrict__ vposed) {
  const int lane = threadIdx.x & 31;
  const int wg   = blockIdx.x * 8 + (threadIdx.x >> 5);   // 5168 waves = 16 * 323
  const int m_group = wg / NT_GROUPS;                     // 0..15 (32 rows each)
  const int n_group = wg % NT_GROUPS;
  const int half = lane >> 4;
  const int lm   = lane & 15;
  const int m  = m_group * 32 + lm;       // A row for M-tile 0; tile 1 is m+16
  const int n0 = n_group * 64;

  int nc0, nc1, nc2, nc3;
  {
    const int a0 = n0 + lm, a1 = n0 + 16 + lm, a2 = n0 + 32 + lm, a3 = n0 + 48 + lm;
    nc0 = (a0 < NCOL) ? a0 : (NCOL - 1);  // clamp: lanes with invalid n never stored
    nc1 = (a1 < NCOL) ? a1 : (NCOL - 1);
    nc2 = (a2 < NCOL) ? a2 : (NCOL - 1);
    nc3 = (a3 < NCOL) ? a3 : (NCOL - 1);
  }
  const float* arow0 = apack + (size_t)m * KPAD + 2 * half;  // A reg r: k = k0 + r + 2*half
  const float* arow1 = arow0 + (size_t)16 * KPAD;
  v8f c00 = {}, c01 = {}, c02 = {}, c03 = {};
  v8f c10 = {}, c11 = {}, c12 = {}, c13 = {};

#define WMMA_STEP(A0, A1, B0, B1, B2, B3)                                              \
  do {                                                                                 \
    c00 = __builtin_amdgcn_wmma_f32_16x16x4_f32(false, (A0), false, (B0), (short)0, c00, false, false); \
    c01 = __builtin_amdgcn_wmma_f32_16x16x4_f32(false, (A0), false, (B1), (short)0, c01, false, false); \
    c02 = __builtin_amdgcn_wmma_f32_16x16x4_f32(false, (A0), false, (B2), (short)0, c02, false, false); \
    c03 = __builtin_amdgcn_wmma_f32_16x16x4_f32(false, (A0), false, (B3), (short)0, c03, false, false); \
    c10 = __builtin_amdgcn_wmma_f32_16x16x4_f32(false, (A1), false, (B0), (short)0, c10, false, false); \
    c11 = __builtin_amdgcn_wmma_f32_16x16x4_f32(false, (A1), false, (B1), (short)0, c11, false, false); \
    c12 = __builtin_amdgcn_wmma_f32_16x16x4_f32(false, (A1), false, (B2), (short)0, c12, false, false); \
    c13 = __builtin_amdgcn_wmma_f32_16x16x4_f32(false, (A1), false, (B3), (short)0, c13, false, false); \
  } while (0)

  // --- region 1: shapedirs, k = 0..11 (sdT rows 10..11 are zero) ---
#pragma unroll
  for (int k0 = 0; k0 < 12; k0 += 4) {
    const v2f a0 = *(const v2f*)(arow0 + k0);              // aligned b64 loads
    const v2f a1 = *(const v2f*)(arow1 + k0);
    const float* b0 = sdt + (size_t)(k0 + 2 * half) * NCOL;
    const v2f bf0 = { b0[nc0], b0[NCOL + nc0] };
    const v2f bf1 = { b0[nc1], b0[NCOL + nc1] };
    const v2f bf2 = { b0[nc2], b0[NCOL + nc2] };
    const v2f bf3 = { b0[nc3], b0[NCOL + nc3] };
    WMMA_STEP(a0, a1, bf0, bf1, bf2, bf3);
  }

  // --- region 2: posedirs, A k = 12..215 (51 steps, all rows in range) ---
  const float* bp0 = pdirs + (size_t)(2 * half) * NCOL + nc0;  // named ptrs: keep global AS
  const float* bp1 = pdirs + (size_t)(2 * half) * NCOL + nc1;
  const float* bp2 = pdirs + (size_t)(2 * half) * NCOL + nc2;
  const float* bp3 = pdirs + (size_t)(2 * half) * NCOL + nc3;
  for (int kk = 0; kk < 204; kk += 4) {                    // rows kk+2h+1 <= 203 < 207
    const v2f a0 = *(const v2f*)(arow0 + 12 + kk);
    const v2f a1 = *(const v2f*)(arow1 + 12 + kk);
    const v2f bf0 = { bp0[0], bp0[NCOL] };                 // 2nd row via imm offset
    const v2f bf1 = { bp1[0], bp1[NCOL] };
    const v2f bf2 = { bp2[0], bp2[NCOL] };
    const v2f bf3 = { bp3[0], bp3[NCOL] };
    WMMA_STEP(a0, a1, bf0, bf1, bf2, bf3);
    bp0 += 4 * (size_t)NCOL; bp1 += 4 * (size_t)NCOL;
    bp2 += 4 * (size_t)NCOL; bp3 += 4 * (size_t)NCOL;
  }
  // --- peeled tail: kk = 204, A k = 216..219 (k=219 is zero in apack) ---
  {
    const v2f a0 = *(const v2f*)(arow0 + 12 + 204);
    const v2f a1 = *(const v2f*)(arow1 + 12 + 204);
    const size_t off1 = half ? 0 : (size_t)NCOL;           // half=1: row 207 -> reuse 206
    const v2f bf0 = { bp0[0], bp0[off1] };
    const v2f bf1 = { bp1[0], bp1[off1] };
    const v2f bf2 = { bp2[0], bp2[off1] };
    const v2f bf3 = { bp3[0], bp3[off1] };
    WMMA_STEP(a0, a1, bf0, bf1, bf2, bf3);
  }
#undef WMMA_STEP

  // --- store: D VGPR r holds row r (+8 for lanes 16..31), col = lane&15 ---
  v8f* accs[8] = { &c00, &c01, &c02, &c03, &c10, &c11, &c12, &c13 };
#pragma unroll
  for (int mt2 = 0; mt2 < 2; ++mt2) {
#pragma unroll
    for (int t = 0; t < 4; ++t) {
      const int n = n0 + t * 16 + lm;
      if (n < NCOL) {
        const float vt = vtmpl[n];
        const v8f acc = *accs[mt2 * 4 + t];
#pragma unroll
        for (int r = 0; r < 8; ++r) {
          const int row = m_group * 32 + mt2 * 16 + 8 * half + r;
          vposed[(size_t)row * NCOL + n] = vt + acc[r];
        }
      }
    }
  }
}

// ---------------------------------------------------------------------------
// Kernel 5: fused LBS + global transform, in-place on d_out.
//   out[b,v,:] = sum_j wT[j,v] * (M_{b,j} @ v_posed[b,v] + t_{b,j})
// ---------------------------------------------------------------------------
__global__ void k_lbs(const float* __restrict__ wt, const float* __restrict__ mt,
                      float* __restrict__ out) {
  const int b = blockIdx.y;
  const int v = blockIdx.x * 256 + threadIdx.x;
  if (v >= V_N) return;
  float* p = out + ((size_t)b * V_N + v) * 3;
  const float px = p[0], py = p[1], pz = p[2];
  float ax = 0.0f, ay = 0.0f, az = 0.0f;
  const float* M = mt + (size_t)b * J_N * 12;   // uniform per block -> scalar loads
#pragma unroll
  for (int j = 0; j < J_N; ++j) {
    const float w = wt[j * V_N + v];            // coalesced
    const float* mj = M + j * 12;
    ax += w * (mj[0] * px + mj[1] * py + mj[2] * pz + mj[9]);
    ay += w * (mj[3] * px + mj[4] * py + mj[5] * pz + mj[10]);
    az += w * (mj[6] * px + mj[7] * py + mj[8] * pz + mj[11]);
  }
  p[0] = ax; p[1] = ay; p[2] = az;
}

// ---------------------------------------------------------------------------
extern "C" void kernel_launch(void* const* d_in, const int* in_sizes, int n_in,
                              void* d_out, int out_size, void* d_ws, size_t ws_size,
                              hipStream_t stream) {
  (void)in_sizes; (void)n_in; (void)out_size; (void)ws_size;
  const float* poses   = (const float*)d_in[0];
  const float* shapes  = (const float*)d_in[1];
  const float* Rh      = (const float*)d_in[2];
  const float* Th      = (const float*)d_in[3];
  const float* vtmpl   = (const float*)d_in[4];
  const float* sdirs   = (const float*)d_in[5];
  const float* pdirs   = (const float*)d_in[6];
  const float* jreg    = (const float*)d_in[7];
  const float* weights = (const float*)d_in[8];
  float* out = (float*)d_out;

  float* apack  = (float*)d_ws;                 // 512*220
  float* mt     = apack + B_N * KPAD;           // 512*24*12
  float* jsmall = mt + B_N * J_N * 12;          // 792
  float* wt     = jsmall + 24 * 3 * 11;         // 24*6890
  float* sdt    = wt + J_N * V_N;               // 12*20670  (total ~2.7 MB)

  k_jsmall<<<dim3(24 * 3 * 11), dim3(256), 0, stream>>>(jreg, sdirs, vtmpl, jsmall);
  k_wtrans<<<dim3((V_N * J_N + 255) / 256), dim3(256), 0, stream>>>(weights, wt);
  k_sdt   <<<dim3((12 * NCOL + 255) / 256), dim3(256), 0, stream>>>(sdirs, sdt);
  k_chain <<<dim3(B_N), dim3(32), 0, stream>>>(poses, shapes, Rh, Th, jsmall, apack, mt);
  k_vposed<<<dim3((NM_GROUPS * NT_GROUPS) / 8), dim3(256), 0, stream>>>(apack, sdt, pdirs, vtmpl, out);
  k_lbs   <<<dim3((V_N + 255) / 256, B_N), dim3(256), 0, stream>>>(wt, mt, out);
}